// MetaSVMNetworks_661424963693
// MI455X (gfx1250) — compile-verified
//
#include <hip/hip_runtime.h>
#include <hip/hip_bf16.h>

// ---------------------------------------------------------------------------
// MetaSVM pipeline for MI455X (gfx1250), all matrix math on V_WMMA_F32_16X16X4_F32.
// Shapes: S[400,3072], Q[240,3072], W[3072,640] -> K[400,640], Fx[240,640]
//         aul = K K^T + 50 I (400x400); X = aul^{-1} P (16 RHS);
//         wk = X^T K (16x640); out = -gamma * Fx wk^T (240x16).
// Problem is HBM-bound (~16 MB in, 2.7 GFLOP) -> keep f32 end-to-end.
// ---------------------------------------------------------------------------

typedef __attribute__((ext_vector_type(2))) float v2f;
typedef __attribute__((ext_vector_type(8))) float v8f;

static __device__ __forceinline__ v8f wmma4(v2f a, v2f b, v8f c) {
  // D = A(16x4) x B(4x16) + C(16x16), f32, wave32.
  return __builtin_amdgcn_wmma_f32_16x16x4_f32(
      /*neg_a=*/false, a, /*neg_b=*/false, b,
      /*c_mod=*/(short)0, c, /*reuse_a=*/false, /*reuse_b=*/false);
}

// ---- Kernel 1: [K ; Fx] = [S ; Q] @ W  (M tiles 0..24 -> K, 25..39 -> Fx) ----
__global__ void k_gemm_feat(const float* __restrict__ S, const float* __restrict__ Q,
                            const float* __restrict__ W, float* __restrict__ Kmat,
                            float* __restrict__ Fx) {
  int wid = (blockIdx.x * blockDim.x + threadIdx.x) >> 5;
  if (wid >= 40 * 10) return;                 // 40 M-tiles x 10 groups of 4 N-tiles
  int lane = threadIdx.x & 31, ln = lane & 15, h = lane >> 4;
  int mt = wid / 10, n0 = (wid % 10) * 64;

  const float* arow = (mt < 25) ? (S + (size_t)(mt * 16 + ln) * 3072)
                                : (Q + (size_t)((mt - 25) * 16 + ln) * 3072);
  const float* wb = W + n0 + ln;              // B[k][n] = W[k*640 + n]
  v8f c0 = {}, c1 = {}, c2 = {}, c3 = {};

  for (int k = 0; k < 3072; k += 4) {
    v2f a = *(const v2f*)(arow + k + 2 * h);  // A[m][2h], A[m][2h+1]
    int r0 = (k + 2 * h) * 640;
    v2f b0 = { wb[r0 +  0], wb[r0 + 640 +  0] };
    v2f b1 = { wb[r0 + 16], wb[r0 + 640 + 16] };
    v2f b2 = { wb[r0 + 32], wb[r0 + 640 + 32] };
    v2f b3 = { wb[r0 + 48], wb[r0 + 640 + 48] };
    c0 = wmma4(a, b0, c0);
    c1 = wmma4(a, b1, c1);
    c2 = wmma4(a, b2, c2);
    c3 = wmma4(a, b3, c3);
  }

  float* out = (mt < 25) ? (Kmat + (size_t)(mt * 16) * 640)
                         : (Fx + (size_t)((mt - 25) * 16) * 640);
#pragma unroll
  for (int r = 0; r < 8; ++r) {
    int row = r + 8 * h;                      // C: c[r] = C[r+8h][ln]
    out[row * 640 + n0 +  0 + ln] = c0[r];
    out[row * 640 + n0 + 16 + ln] = c1[r];
    out[row * 640 + n0 + 32 + ln] = c2[r];
    out[row * 640 + n0 + 48 + ln] = c3[r];
  }
}

// ---- Kernel 2: AUG[:,0:400] = K K^T + 50 I  (AUG stride 416, RHS cols later) ----
__global__ void k_syrk(const float* __restrict__ Kmat, float* __restrict__ AUG) {
  int wid = (blockIdx.x * blockDim.x + threadIdx.x) >> 5;
  if (wid >= 625) return;                     // 25 x 25 tiles
  int lane = threadIdx.x & 31, ln = lane & 15, h = lane >> 4;
  int mt = wid / 25, nt = wid % 25;
  const float* ra = Kmat + (size_t)(mt * 16 + ln) * 640;
  const float* rb = Kmat + (size_t)(nt * 16 + ln) * 640;  // B[k][n] = K[n][k]
  v8f c = {};
  for (int k = 0; k < 640; k += 4) {
    v2f a = *(const v2f*)(ra + k + 2 * h);
    v2f b = *(const v2f*)(rb + k + 2 * h);
    c = wmma4(a, b, c);
  }
#pragma unroll
  for (int r = 0; r < 8; ++r) {
    int row = mt * 16 + r + 8 * h, col = nt * 16 + ln;
    AUG[(size_t)row * 416 + col] = c[r] + (row == col ? 50.0f : 0.0f);
  }
}

// ---- Kernel 3: blocked Gauss-Jordan solve of AUG x = P, 16-wide pivots, WMMA ----
__global__ void __launch_bounds__(512) k_solve(float* __restrict__ AUG) {
  __shared__ float sM[16][33];     // [D | I] Jordan workspace for 16x16 inverse
  __shared__ float sDinv[16][17];
  __shared__ float sU[16][404];    // NEGATED normalized pivot row panel (cols >= base+16)
  int tid = threadIdx.x;
  int lane = tid & 31, ln = lane & 15, h = lane >> 4, wv = tid >> 5;  // 16 waves

  // RHS init: AUG[i][400+b] = (i/25 == b)
  for (int idx = tid; idx < 400 * 16; idx += 512) {
    int i = idx >> 4, b = idx & 15;
    AUG[(size_t)i * 416 + 400 + b] = ((i / 25) == b) ? 1.0f : 0.0f;
  }
  __syncthreads();

  for (int jb = 0; jb < 25; ++jb) {
    int base = jb * 16;
    int ncols = 25 - jb;                       // block-cols at base+16 (incl. RHS block)

    // (a) invert 16x16 diagonal block in LDS (each thread owns one of 16x32 cells)
    {
      int r = tid >> 5, cc = tid & 31;
      sM[r][cc] = (cc < 16) ? AUG[(size_t)(base + r) * 416 + base + cc]
                            : ((cc - 16 == r) ? 1.0f : 0.0f);
      __syncthreads();
      for (int p = 0; p < 16; ++p) {
        float pv = sM[p][p];
        float pr = sM[p][cc];
        float f  = sM[r][p];
        float cur = sM[r][cc];
        __syncthreads();
        float prn = pr / pv;
        sM[r][cc] = (r == p) ? prn : (cur - f * prn);
        __syncthreads();
      }
      if (cc >= 16) sDinv[r][cc - 16] = sM[r][cc];
    }
    __syncthreads();

    // (b) pivot block-row normalize: NewRow = Dinv @ OldRow (cols >= base+16);
    //     store +NewRow to AUG, -NewRow to LDS (f32 WMMA has no A-negate).
    for (int ct = wv; ct < ncols; ct += 16) {
      int col0 = base + 16 + ct * 16;
      v8f c = {};
#pragma unroll
      for (int kk = 0; kk < 16; kk += 4) {
        v2f a = { sDinv[ln][kk + 2 * h], sDinv[ln][kk + 2 * h + 1] };
        const float* bp = AUG + (size_t)(base + kk + 2 * h) * 416 + col0 + ln;
        v2f b = { bp[0], bp[416] };
        c = wmma4(a, b, c);
      }
#pragma unroll
      for (int r = 0; r < 8; ++r) {
        int rr = r + 8 * h;
        AUG[(size_t)(base + rr) * 416 + col0 + ln] = c[r];
        sU[rr][ct * 16 + ln] = -c[r];
      }
    }
    __syncthreads();

    // (c) eliminate all other block-rows: C += L @ (-U) via WMMA rank-16 updates
    int total = 24 * ncols;
    for (int job = wv; job < total; job += 16) {
      int rb = job / ncols, ct = job % ncols;
      int ib = rb + (rb >= jb);                // skip pivot block-row
      int col0 = base + 16 + ct * 16;
      const float* lrow = AUG + (size_t)(ib * 16 + ln) * 416 + base;
      float* crow = AUG + (size_t)(ib * 16) * 416 + col0 + ln;
      v8f c;
#pragma unroll
      for (int r = 0; r < 8; ++r) c[r] = crow[(size_t)(r + 8 * h) * 416];
#pragma unroll
      for (int kk = 0; kk < 16; kk += 4) {
        v2f a = *(const v2f*)(lrow + kk + 2 * h);
        v2f b = { sU[kk + 2 * h][ct * 16 + ln], sU[kk + 2 * h + 1][ct * 16 + ln] };
        c = wmma4(a, b, c);
      }
#pragma unroll
      for (int r = 0; r < 8; ++r) crow[(size_t)(r + 8 * h) * 416] = c[r];
    }
    __syncthreads();
  }
  // AUG[:,400:416] now holds X = aul^{-1} P.
}

// ---- Kernel 4: wk[16,640] = X^T @ K  (A[m][k] = AUG[k][400+m]) ----
__global__ void k_wk(const float* __restrict__ AUG, const float* __restrict__ Kmat,
                     float* __restrict__ wk) {
  int wid = (blockIdx.x * blockDim.x + threadIdx.x) >> 5;
  if (wid >= 40) return;
  int lane = threadIdx.x & 31, ln = lane & 15, h = lane >> 4;
  int col0 = wid * 16;
  v8f c = {};
  for (int k = 0; k < 400; k += 4) {
    int kr = k + 2 * h;
    v2f a = { AUG[(size_t)kr * 416 + 400 + ln], AUG[(size_t)(kr + 1) * 416 + 400 + ln] };
    v2f b = { Kmat[(size_t)kr * 640 + col0 + ln], Kmat[(size_t)(kr + 1) * 640 + col0 + ln] };
    c = wmma4(a, b, c);
  }
#pragma unroll
  for (int r = 0; r < 8; ++r)
    wk[(size_t)(r + 8 * h) * 640 + col0 + ln] = c[r];
}

// ---- Kernel 5: out[240,16] = -gamma * Fx @ wk^T ----
__global__ void k_logits(const float* __restrict__ Fx, const float* __restrict__ wk,
                         const float* __restrict__ gamma, float* __restrict__ out) {
  int wid = (blockIdx.x * blockDim.x + threadIdx.x) >> 5;
  if (wid >= 15) return;
  int lane = threadIdx.x & 31, ln = lane & 15, h = lane >> 4;
  const float* ra = Fx + (size_t)(wid * 16 + ln) * 640;
  const float* rb = wk + (size_t)ln * 640;    // B[k][n] = wk[n][k]
  v8f c = {};
  for (int k = 0; k < 640; k += 4) {
    v2f a = *(const v2f*)(ra + k + 2 * h);
    v2f b = *(const v2f*)(rb + k + 2 * h);
    c = wmma4(a, b, c);
  }
  float g = -gamma[0];
#pragma unroll
  for (int r = 0; r < 8; ++r)
    out[(size_t)(wid * 16 + r + 8 * h) * 16 + ln] = g * c[r];
}

extern "C" void kernel_launch(void* const* d_in, const int* in_sizes, int n_in,
                              void* d_out, int out_size, void* d_ws, size_t ws_size,
                              hipStream_t stream) {
  const float* S     = (const float*)d_in[0];   // [16,25,3,32,32] -> [400,3072]
  const float* Q     = (const float*)d_in[1];   // [16,15,3,32,32] -> [240,3072]
  const float* W     = (const float*)d_in[2];   // [3072,640]
  const float* gamma = (const float*)d_in[3];   // [1]
  float* ws   = (float*)d_ws;
  float* Kmat = ws;                 // 400*640   = 256000
  float* Fx   = ws + 256000;        // 240*640   = 153600
  float* AUG  = ws + 409600;        // 400*416   = 166400 (aul | X)
  float* wk   = ws + 576000;        // 16*640    = 10240  (total ~2.35 MB)
  float* out  = (float*)d_out;      // [240,16] f32

  hipLaunchKernelGGL(k_gemm_feat, dim3(50), dim3(256), 0, stream, S, Q, W, Kmat, Fx);
  hipLaunchKernelGGL(k_syrk,      dim3(79), dim3(256), 0, stream, Kmat, AUG);
  hipLaunchKernelGGL(k_solve,     dim3(1),  dim3(512), 0, stream, AUG);
  hipLaunchKernelGGL(k_wk,        dim3(5),  dim3(256), 0, stream, AUG, Kmat, wk);
  hipLaunchKernelGGL(k_logits,    dim3(2),  dim3(256), 0, stream, Fx, wk, gamma, out);
}